// linear_cla_19189913879290
// MI455X (gfx1250) — compile-verified
//
#include <hip/hip_runtime.h>

// ---------------------------------------------------------------------------
// linear-CLA style fused kernel for MI455X (gfx1250, wave32)
//
//   b=16, c=512, h=w=64, hw=4096
//   Pass 1: partial hw-sums of x via V_WMMA_F32_16X16X4_F32 (B = ones),
//           8-way hw split -> 4096 waves for latency hiding
//   Pass 1b: fold 8 partials -> y[b,c] = mean
//   Pass 2: 5-tap channel conv -> Qf,Kf; z_new = z + Kf; f = Qf*(1/(Qf*(zn+eps)))
//   Pass 3: depthwise 3x3 conv from an async-staged LDS tile (66x66, zero halo);
//           s_new = s + Kf*V ; out = f * s_new.
//           s / s_new / out use non-temporal hints so x (128 MB) stays resident
//           in the 192 MB L2 between pass 1 and pass 3.
//
// HBM roofline: ~512 MB effective traffic @ 23.3 TB/s -> ~22 us. FLOPs trivial.
// ---------------------------------------------------------------------------

typedef __attribute__((ext_vector_type(2))) float v2f;
typedef __attribute__((ext_vector_type(8))) float v8f;

#define BB   16
#define CC   512
#define HH   64
#define WW   64
#define HWSZ 4096
#define NSEG 8
#define SEGW (HWSZ / NSEG)   // 512 hw positions per segment
#define EPSF 1e-6f

// ---------------------------------------------------------------------------
// Kernel 1: partial sums over one hw segment for a 16-channel tile using
// V_WMMA_F32_16X16X4_F32.  A[m][k] = x[b, c0+m, hw0+k], B = all-ones 4x16:
// every column of D holds sum_hw x, so B's register layout is irrelevant.
// A layout (ISA 7.12.2): lane = m + 16*(k>=2), vgpr = k&1.
// D layout: vgpr r, lanes 0-15 -> row r; lanes 16-31 -> row r+8.
// One wave per block; EXEC all-ones around every WMMA.
// ---------------------------------------------------------------------------
__global__ __launch_bounds__(32) void k_mean_wmma(const float* __restrict__ x,
                                                  float* __restrict__ part) {
    const int b    = blockIdx.x >> 8;            // 16 batches
    const int rem  = blockIdx.x & 255;
    const int c0   = (rem >> 3) << 4;            // 32 tiles of 16 channels
    const int seg  = rem & (NSEG - 1);           // 8 hw segments
    const int lane = threadIdx.x;                // 0..31 (wave32)
    const int m    = lane & 15;
    const int kh   = lane >> 4;                  // 0 -> K{0,1}, 1 -> K{2,3}

    const float* row =
        x + (size_t)(b * CC + c0 + m) * HWSZ + seg * SEGW + 2 * kh;

    v2f bones; bones.x = 1.0f; bones.y = 1.0f;
    v8f acc = {};
    for (int hw0 = 0; hw0 < SEGW; hw0 += 4) {
        v2f a;
        a.x = row[hw0];
        a.y = row[hw0 + 1];
        acc = __builtin_amdgcn_wmma_f32_16x16x4_f32(
            /*neg_a=*/false, a, /*neg_b=*/false, bones,
            /*c_mod=*/(short)0, acc, /*reuse_a=*/false, /*reuse_b=*/false);
    }

    // part layout: [b*CC + c][seg]
    if (lane == 0) {
#pragma unroll
        for (int r = 0; r < 8; ++r)
            part[(size_t)((b * CC + c0 + r) << 3) + seg] = acc[r];
    } else if (lane == 16) {
#pragma unroll
        for (int r = 0; r < 8; ++r)
            part[(size_t)((b * CC + c0 + 8 + r) << 3) + seg] = acc[r];
    }
}

// ---------------------------------------------------------------------------
// Kernel 1b: fold the 8 partials into the channel mean.
// ---------------------------------------------------------------------------
__global__ __launch_bounds__(256) void k_reduce(const float* __restrict__ part,
                                                float* __restrict__ y) {
    const int idx = blockIdx.x * 256 + threadIdx.x;   // b*CC + c  (0..8191)
    float acc = 0.0f;
#pragma unroll
    for (int sgi = 0; sgi < NSEG; ++sgi) acc += part[(size_t)(idx << 3) + sgi];
    y[idx] = acc * (1.0f / (float)HWSZ);
}

// ---------------------------------------------------------------------------
// Kernel 2: channel-axis 5-tap convs (SAME, zero pad), elu+1 gates,
// z_new, and the fused output scale f = Qf * (1/(Qf*(z_new+eps))).
// grid = 16 (batch), block = 512 (channel).
// ---------------------------------------------------------------------------
__global__ __launch_bounds__(512) void k_gate(const float* __restrict__ y,
                                              const float* __restrict__ z,
                                              const float* __restrict__ wq,
                                              const float* __restrict__ wk,
                                              float* __restrict__ Kf_out,
                                              float* __restrict__ f_out,
                                              float* __restrict__ znew_out) {
    const int b = blockIdx.x;
    const int c = threadIdx.x;

    float yv[5];
#pragma unroll
    for (int i = 0; i < 5; ++i) {
        const int cc = c - 2 + i;
        yv[i] = (cc >= 0 && cc < CC) ? y[b * CC + cc] : 0.0f;
    }
    float q = 0.0f, k = 0.0f;
#pragma unroll
    for (int i = 0; i < 5; ++i) {
        q += wq[i] * yv[i];
        k += wk[i] * yv[i];
    }
    // elu(t)+1 : t>0 ? t+1 : exp(t)
    const float Qf = (q > 0.0f) ? (q + 1.0f) : expf(q);
    const float Kf = (k > 0.0f) ? (k + 1.0f) : expf(k);

    const float zn = z[b * CC + c] + Kf;            // z_new
    const float qz = 1.0f / (Qf * (zn + EPSF));     // matches reference order
    Kf_out[b * CC + c]   = Kf;
    f_out[b * CC + c]    = Qf * qz;
    znew_out[b * CC + c] = zn;
}

// ---------------------------------------------------------------------------
// Kernel 3: one block per (b,c) image. Stage the 64x64 channel image into a
// 66x66 LDS tile (zero halo) with gfx1250 async LDS loads, then depthwise 3x3
// conv + s_new + out in a single streaming pass.
// tile[] is the only __shared__ object -> LDS offset of tile[0] is 0, so the
// byte index doubles as the wave-relative LDS address for the async engine.
// s / s_new / out are non-temporal (read/write once) so x stays hot in L2.
// ---------------------------------------------------------------------------
__global__ __launch_bounds__(256) void k_main(const float* __restrict__ x,
                                              const float* __restrict__ s,
                                              const float* __restrict__ wv,
                                              const float* __restrict__ Kf_in,
                                              const float* __restrict__ f_in,
                                              float* __restrict__ out,
                                              float* __restrict__ s_new) {
    __shared__ float tile[66 * 66];

    const int bc  = blockIdx.x;        // b*512 + c
    const int c   = bc & (CC - 1);
    const int tid = threadIdx.x;

    // Zero the halo border (interior is fully overwritten by the async loads).
    if (tid < 66) {
        tile[tid]           = 0.0f;    // row 0
        tile[65 * 66 + tid] = 0.0f;    // row 65
    }
    if (tid < 64) {
        tile[(tid + 1) * 66]      = 0.0f;  // col 0
        tile[(tid + 1) * 66 + 65] = 0.0f;  // col 65
    }

    // Async-copy the 64x64 interior (row stride 66 in LDS), 16 B per lane
    // per issue; 4 issues x 256 threads covers 4096 floats.
    const unsigned long long sbase = (unsigned long long)(x + (size_t)bc * HWSZ);
#pragma unroll
    for (int it = 0; it < 4; ++it) {
        const int chunk      = tid + it * 256;                 // 0..1023
        const int r          = chunk >> 4;
        const int col4       = (chunk & 15) << 2;
        const unsigned ldsad = (unsigned)(((r + 1) * 66 + 1 + col4) * 4);
        const unsigned goff  = (unsigned)((r * WW + col4) * 4);
        asm volatile("global_load_async_to_lds_b128 %0, %1, %2"
                     :
                     : "v"(ldsad), "v"(goff), "s"(sbase)
                     : "memory");
    }
    asm volatile("s_wait_asynccnt 0x0" ::: "memory");
    __syncthreads();

    // Per-channel depthwise weights + gates (uniform -> scalar loads).
    const float w00 = wv[c * 9 + 0], w01 = wv[c * 9 + 1], w02 = wv[c * 9 + 2];
    const float w10 = wv[c * 9 + 3], w11 = wv[c * 9 + 4], w12 = wv[c * 9 + 5];
    const float w20 = wv[c * 9 + 6], w21 = wv[c * 9 + 7], w22 = wv[c * 9 + 8];
    const float Kf = Kf_in[bc];
    const float f  = f_in[bc];

    const size_t base = (size_t)bc * HWSZ;
#pragma unroll 4
    for (int it = 0; it < 16; ++it) {
        const int p   = it * 256 + tid;      // coalesced across the block
        const int r   = p >> 6;
        const int col = p & 63;
        const float* t = &tile[r * 66 + col];
        const float v = w00 * t[0]   + w01 * t[1]   + w02 * t[2]
                      + w10 * t[66]  + w11 * t[67]  + w12 * t[68]
                      + w20 * t[132] + w21 * t[133] + w22 * t[134];
        const float sv = __builtin_nontemporal_load(&s[base + p]) + Kf * v;
        __builtin_nontemporal_store(sv, &s_new[base + p]);
        __builtin_nontemporal_store(f * sv, &out[base + p]);
    }
}

// ---------------------------------------------------------------------------
extern "C" void kernel_launch(void* const* d_in, const int* in_sizes, int n_in,
                              void* d_out, int out_size, void* d_ws, size_t ws_size,
                              hipStream_t stream) {
    const float* x  = (const float*)d_in[0];   // [16,512,64,64]
    const float* s  = (const float*)d_in[1];   // [16,512,1,4096]
    const float* z  = (const float*)d_in[2];   // [16,1,512,1]
    const float* wq = (const float*)d_in[3];   // [1,1,5]
    const float* wk = (const float*)d_in[4];   // [1,1,5]
    const float* wv = (const float*)d_in[5];   // [512,1,3,3]

    const size_t NOUT = (size_t)BB * CC * HWSZ;     // 33,554,432
    float* out   = (float*)d_out;                   // [b,c,h,w]
    float* s_new = out + NOUT;                      // [b,c,1,hw]
    float* z_new = s_new + NOUT;                    // [b,1,c,1]

    float* part = (float*)d_ws;                     // [b*c][8] partial sums
    float* y    = part + (size_t)BB * CC * NSEG;    // [b,c] means
    float* Kf   = y + BB * CC;                      // [b,c]
    float* f    = Kf + BB * CC;                     // [b,c] fused out scale

    k_mean_wmma<<<BB * 32 * NSEG, 32, 0, stream>>>(x, part);
    k_reduce<<<BB * CC / 256, 256, 0, stream>>>(part, y);
    k_gate<<<BB, CC, 0, stream>>>(y, z, wq, wk, Kf, f, z_new);
    k_main<<<BB * CC, 256, 0, stream>>>(x, s, wv, Kf, f, out, s_new);
}